// HungarianMatcher_16947940950496
// MI455X (gfx1250) — compile-verified
//
#include <hip/hip_runtime.h>

// DETR HungarianMatcher cost matrix, CDNA5 (gfx1250, wave32).
//
// cost_class is computed as an exact f32 WMMA matmul against a
// lane-synthesized negative one-hot matrix:
//   cost_class[b] = pred_logits[b] (Q x NC) @ (-onehot(labels[b])^T) (NC x T)
// using V_WMMA_F32_16X16X4_F32, accumulating K=92 in 23 steps of 4.
// bbox-L1 and GIoU terms are fused in the epilogue on the D-matrix layout.

typedef float v2f __attribute__((ext_vector_type(2)));
typedef float v8f __attribute__((ext_vector_type(8)));

namespace {
constexpr int kB  = 128;
constexpr int kQ  = 900;
constexpr int kT  = 100;
constexpr int kNC = 92;
constexpr int kTT = (kT + 15) / 16;   // 7 T-tiles of 16 columns
constexpr float kCostClass = 1.0f;
constexpr float kCostBbox  = 5.0f;
constexpr float kCostGiou  = 2.0f;
}

__global__ __launch_bounds__(32)
void detr_matcher_cost_wmma(const float* __restrict__ pred_logits,   // [B,Q,NC]
                            const float* __restrict__ pred_boxes,    // [B,Q,4] cxcywh
                            const float* __restrict__ target_boxes,  // [B,T,4] cxcywh
                            const int*   __restrict__ target_labels, // [B,T]
                            float* __restrict__ out)                 // [B,Q,T]
{
  const int lane = threadIdx.x;      // 0..31 (wave32)
  const int lo   = lane & 15;
  const int hi   = lane >> 4;        // 0 or 1
  const int b    = blockIdx.y;
  const int q0   = blockIdx.x * 16;  // Q-tile base row

  // ---- Per-lane labels for each T-tile (B-matrix column n == lo). ----
  // Out-of-range columns get label=-1 (never matches any class -> 0 column).
  int lab[kTT];
#pragma unroll
  for (int tt = 0; tt < kTT; ++tt) {
    const int t = tt * 16 + lo;
    lab[tt] = (t < kT) ? target_labels[(size_t)b * kT + t] : -1;
  }

  // ---- Accumulators: one 16x16 f32 tile per T-tile. ----
  const v8f vzero = {0.f, 0.f, 0.f, 0.f, 0.f, 0.f, 0.f, 0.f};
  v8f acc[kTT];
#pragma unroll
  for (int tt = 0; tt < kTT; ++tt) acc[tt] = vzero;

  // A-fragment (16x4 f32): lanes 0-15 hold M=lo,K={kk,kk+1};
  // lanes 16-31 hold M=lo,K={kk+2,kk+3}.  8-byte aligned float2 loads.
  int arow = q0 + lo;
  if (arow > kQ - 1) arow = kQ - 1;
  const float* aptr = pred_logits + ((size_t)b * kQ + arow) * kNC + 2 * hi;

#pragma unroll
  for (int kk = 0; kk < kNC; kk += 4) {   // 23 K-steps (92 = 23*4 exactly)
    const v2f a  = *(const v2f*)(aptr + kk);
    const int k0 = kk + 2 * hi;
#pragma unroll
    for (int tt = 0; tt < kTT; ++tt) {
      // Negative one-hot B-fragment (4x16 f32), synthesized in-register:
      // lanes 0-15: rows K=kk,kk+1 ; lanes 16-31: rows K=kk+2,kk+3 ; N=lo.
      v2f bf;
      bf[0] = (lab[tt] == k0    ) ? -1.0f : 0.0f;
      bf[1] = (lab[tt] == k0 + 1) ? -1.0f : 0.0f;
      acc[tt] = __builtin_amdgcn_wmma_f32_16x16x4_f32(
          /*neg_a=*/false, a, /*neg_b=*/false, bf,
          /*c_mod=*/(short)0, acc[tt], /*reuse_a=*/false, /*reuse_b=*/false);
    }
  }
  // acc[tt][i] now holds cost_class for (m = q0 + i + 8*hi, t = tt*16 + lo).

  // ---- Target-box values per T-tile, per lane (column n = lo fixed). ----
  float tcx[kTT], tcy[kTT], tw[kTT], th[kTT];
  float tx1[kTT], ty1[kTT], tx2[kTT], ty2[kTT], tA[kTT];
#pragma unroll
  for (int tt = 0; tt < kTT; ++tt) {
    const int t  = tt * 16 + lo;
    const int tc = (t < kT) ? t : kT - 1;
    const float4 tb = *(const float4*)(target_boxes + ((size_t)b * kT + tc) * 4);
    tcx[tt] = tb.x; tcy[tt] = tb.y; tw[tt] = tb.z; th[tt] = tb.w;
    tx1[tt] = tb.x - 0.5f * tb.z;  ty1[tt] = tb.y - 0.5f * tb.w;
    tx2[tt] = tb.x + 0.5f * tb.z;  ty2[tt] = tb.y + 0.5f * tb.w;
    tA[tt]  = (tx2[tt] - tx1[tt]) * (ty2[tt] - ty1[tt]);
  }

  // ---- Fused epilogue: L1 + GIoU + class, then store. ----
#pragma unroll
  for (int i = 0; i < 8; ++i) {
    const int m  = q0 + i + 8 * hi;
    const int mc = (m < kQ) ? m : kQ - 1;
    const float4 pb = *(const float4*)(pred_boxes + ((size_t)b * kQ + mc) * 4);
    const float px1 = pb.x - 0.5f * pb.z, py1 = pb.y - 0.5f * pb.w;
    const float px2 = pb.x + 0.5f * pb.z, py2 = pb.y + 0.5f * pb.w;
    const float pA  = (px2 - px1) * (py2 - py1);
#pragma unroll
    for (int tt = 0; tt < kTT; ++tt) {
      const int t = tt * 16 + lo;
      // L1 on cxcywh
      const float l1 = fabsf(pb.x - tcx[tt]) + fabsf(pb.y - tcy[tt]) +
                       fabsf(pb.z - tw[tt])  + fabsf(pb.w - th[tt]);
      // intersection
      const float ix1 = fmaxf(px1, tx1[tt]), iy1 = fmaxf(py1, ty1[tt]);
      const float ix2 = fminf(px2, tx2[tt]), iy2 = fminf(py2, ty2[tt]);
      const float iw  = fmaxf(ix2 - ix1, 0.f), ih = fmaxf(iy2 - iy1, 0.f);
      const float inter = iw * ih;
      const float uni   = pA + tA[tt] - inter;
      const float iou   = inter * __builtin_amdgcn_rcpf(uni);
      // enclosing box
      const float ex1 = fminf(px1, tx1[tt]), ey1 = fminf(py1, ty1[tt]);
      const float ex2 = fmaxf(px2, tx2[tt]), ey2 = fmaxf(py2, ty2[tt]);
      const float ew  = fmaxf(ex2 - ex1, 0.f), eh = fmaxf(ey2 - ey1, 0.f);
      const float enc = ew * eh;
      const float giou = iou - (enc - uni) * __builtin_amdgcn_rcpf(enc);
      // acc already carries the minus sign of cost_class (B = -onehot)
      const float c = kCostBbox * l1 + kCostClass * acc[tt][i] - kCostGiou * giou;
      if (m < kQ && t < kT)
        out[((size_t)b * kQ + m) * kT + t] = c;
    }
  }
}

extern "C" void kernel_launch(void* const* d_in, const int* in_sizes, int n_in,
                              void* d_out, int out_size, void* d_ws, size_t ws_size,
                              hipStream_t stream) {
  const float* pred_logits   = (const float*)d_in[0];
  const float* pred_boxes    = (const float*)d_in[1];
  const float* target_boxes  = (const float*)d_in[2];
  const int*   target_labels = (const int*)d_in[3];
  float* out = (float*)d_out;

  dim3 grid((kQ + 15) / 16, kB);  // 57 Q-tiles x 128 batches, 1 wave each
  dim3 block(32);
  detr_matcher_cost_wmma<<<grid, block, 0, stream>>>(
      pred_logits, pred_boxes, target_boxes, target_labels, out);
}